// DilationConv_53429393162850
// MI455X (gfx1250) — compile-verified
//
#include <hip/hip_runtime.h>

// ---------------------------------------------------------------------------
// 3x3 grayscale dilation (max-plus "convolution"), zero-padded, stride 1.
// x: (16,64,256,256) f32, se: (64,3,3) f32 -> out: (16,64,256,256) f32
// Bandwidth-bound (~512MB min traffic -> ~22us @ 23.3TB/s). WMMA cannot
// express max-plus (no max in the WMMA semiring), so the gfx1250-specific
// path used is the async global->LDS data mover (ASYNCcnt) + split waits.
// ROWS=32 tile: halo read overhead 2/32 = 6% of reads (~3% of total traffic).
// ---------------------------------------------------------------------------

#define HAS_ASYNC_B128 __has_builtin(__builtin_amdgcn_global_load_async_to_lds_b128)
#define HAS_WAIT_ASYNC __has_builtin(__builtin_amdgcn_s_wait_asynccnt)

typedef float f4  __attribute__((ext_vector_type(4)));
typedef int   i4  __attribute__((vector_size(16)));      // matches builtin's V4i

typedef __attribute__((address_space(1))) i4* gptr_i4;   // global int4*
typedef __attribute__((address_space(3))) i4* lptr_i4;   // LDS int4*

constexpr int Bb   = 16;
constexpr int Cc   = 64;
constexpr int Hh   = 256;
constexpr int Ww   = 256;
constexpr int ROWS = 32;          // output rows per block
constexpr int LROWS = ROWS + 2;   // staged input rows incl. halo (34)
constexpr int TPB  = 256;         // 8 wave32

__global__ __launch_bounds__(TPB)
void DilationConv_53429393162850_kernel(const float* __restrict__ x,
                                        const float* __restrict__ se,
                                        float* __restrict__ out)
{
    __shared__ __align__(16) float smem[LROWS * Ww];   // 34 KB

    const int tid  = threadIdx.x;
    const int tilesPerPlane = Hh / ROWS;               // 8
    const int bc   = blockIdx.x / tilesPerPlane;       // 0 .. B*C-1
    const int tile = blockIdx.x % tilesPerPlane;
    const int c    = bc % Cc;
    const int row0 = tile * ROWS;

    const float* xplane = x + (size_t)bc * Hh * Ww;

    // ---- stage rows (row0-1 .. row0+ROWS) into LDS; zero the halo ----------
    constexpr int CHUNKS = LROWS * (Ww / 4);           // 34*64 = 2176 float4s
    for (int ch = tid; ch < CHUNKS; ch += TPB) {
        const int r  = ch >> 6;                        // lds row 0..33
        const int c4 = (ch & 63) << 2;                 // column (floats)
        const int gr = row0 - 1 + r;                   // global input row
        float* ldsdst = &smem[r * Ww + c4];
        if (gr >= 0 && gr < Hh) {
            const float* gsrc = xplane + (size_t)gr * Ww + c4;
#if HAS_ASYNC_B128
            // global_load_async_to_lds_b128: 16B per lane, memory -> LDS,
            // no VGPR staging; tracked with ASYNCcnt.
            __builtin_amdgcn_global_load_async_to_lds_b128(
                (gptr_i4)gsrc, (lptr_i4)ldsdst,
                /*imm_offset=*/0, /*cpol=*/0);
#else
            *(f4*)ldsdst = *(const f4*)gsrc;           // fallback: vmem+ds
#endif
        } else {
            f4 z = {0.f, 0.f, 0.f, 0.f};               // zero padding rows
            *(f4*)ldsdst = z;
        }
    }
#if HAS_ASYNC_B128 && HAS_WAIT_ASYNC
    __builtin_amdgcn_s_wait_asynccnt(0);
#endif
    __syncthreads();

    // ---- structuring element for this channel (wave-uniform -> scalar) -----
    const float* sep = se + c * 9;
    const float s00 = sep[0], s01 = sep[1], s02 = sep[2];
    const float s10 = sep[3], s11 = sep[4], s12 = sep[5];
    const float s20 = sep[6], s21 = sep[7], s22 = sep[8];

    const int j0   = (tid & 63) << 2;  // first of 4 output columns
    const int rsub = tid >> 6;         // 0..3: row within a 4-row slice

    const float NEG = -__builtin_inff();

    #pragma unroll
    for (int it = 0; it < ROWS / 4; ++it) {
        const int rl = it * 4 + rsub;              // local output row 0..31
        float a0 = NEG, a1 = NEG, a2 = NEG, a3 = NEG;

        #pragma unroll
        for (int di = 0; di < 3; ++di) {
            const float* rowp = &smem[(rl + di) * Ww];  // lds rows rl..rl+2
            const float sL = (di == 0) ? s00 : (di == 1) ? s10 : s20;
            const float sM = (di == 0) ? s01 : (di == 1) ? s11 : s21;
            const float sR = (di == 0) ? s02 : (di == 1) ? s12 : s22;

            const float vm1 = (j0 == 0)      ? 0.f : rowp[j0 - 1];  // col pad
            const f4    v   = *(const f4*)&rowp[j0];
            const float vp4 = (j0 + 4 == Ww) ? 0.f : rowp[j0 + 4];  // col pad

            a0 = fmaxf(a0, fmaxf(fmaxf(vm1 + sL, v.x + sM), v.y + sR));
            a1 = fmaxf(a1, fmaxf(fmaxf(v.x + sL, v.y + sM), v.z + sR));
            a2 = fmaxf(a2, fmaxf(fmaxf(v.y + sL, v.z + sM), v.w + sR));
            a3 = fmaxf(a3, fmaxf(fmaxf(v.z + sL, v.w + sM), vp4 + sR));
        }

        const size_t oidx = ((size_t)bc * Hh + (row0 + rl)) * Ww + j0;
        f4 o = {a0, a1, a2, a3};
        *(f4*)&out[oidx] = o;                      // global_store_b128
    }
}

extern "C" void kernel_launch(void* const* d_in, const int* in_sizes, int n_in,
                              void* d_out, int out_size, void* d_ws, size_t ws_size,
                              hipStream_t stream) {
    const float* x  = (const float*)d_in[0];   // (16,64,256,256) f32
    const float* se = (const float*)d_in[1];   // (64,3,3) f32
    float* out      = (float*)d_out;           // (16,64,256,256) f32

    const int grid = Bb * Cc * (Hh / ROWS);    // 8192 blocks
    DilationConv_53429393162850_kernel<<<grid, TPB, 0, stream>>>(x, se, out);
}